// DigitCapsule_31885837206082
// MI455X (gfx1250) — compile-verified
//
#include <hip/hip_runtime.h>
#include <hip/hip_bf16.h>

typedef __attribute__((ext_vector_type(2))) float v2f;
typedef __attribute__((ext_vector_type(8))) float v8f;

#define R_NODES 1152
#define IN_CH 8
#define OUT_CH 16
#define DIGITS 10
#define BATCH 256
#define NUM_ITER 3

#define U_RSTRIDE (R_NODES * IN_CH)          /* 9216: u[b,r,i]  b-stride */
#define W_RSTRIDE (DIGITS * OUT_CH * IN_CH)  /* 1280: W[r,d,o,i] r-stride */
#define W_DSTRIDE (OUT_CH * IN_CH)           /* 128 */
#define S_BSTRIDE (DIGITS * OUT_CH)          /* 160: s/v[b,d,o] b-stride */
#define S_ELEMS (BATCH * DIGITS * OUT_CH)    /* 40960 */

#define RC_SIZE 16       /* route nodes per s-accum wave; 72 chunks */
#define BT_PER_BLOCK 4   /* batch tiles (of 16) per agreement wave */

// ---------------------------------------------------------------------------
// c[r,:] = softmax(b_logits[r,:]) over the 10 digits (axis=1 in reference)
// ---------------------------------------------------------------------------
__global__ void softmax_kernel(const float* __restrict__ blog,
                               float* __restrict__ c) {
  int r = blockIdx.x * blockDim.x + threadIdx.x;
  if (r >= R_NODES) return;
  float x[DIGITS];
  float mx = -3.402823e38f;
#pragma unroll
  for (int d = 0; d < DIGITS; ++d) { x[d] = blog[r * DIGITS + d]; mx = fmaxf(mx, x[d]); }
  float sum = 0.0f;
#pragma unroll
  for (int d = 0; d < DIGITS; ++d) { x[d] = __expf(x[d] - mx); sum += x[d]; }
  float inv = 1.0f / sum;
#pragma unroll
  for (int d = 0; d < DIGITS; ++d) c[r * DIGITS + d] = x[d] * inv;
}

// ---------------------------------------------------------------------------
// s[b,d,o] += sum_{r in chunk} c[r,d] * sum_i W[r,d,o,i] * u[b,r,i]
// One wave per (batch-tile of 16, r-chunk of 16). u_hat is recomputed on the
// fly via chained V_WMMA_F32_16X16X4_F32 (K=8 = 2 x K4), c folded into B.
// Fragment layouts per CDNA5 ISA:
//   A 16x4 : lane m=lane&15, VGPR j -> K = 2*(lane>>4)+j   (float2 of u)
//   B 4x16 : lane n=lane&15, VGPR j -> K = 2*(lane>>4)+j   (float2 of W)
//   C 16x16: elem j -> (M = j + 8*(lane>>4), N = lane&15)
// ---------------------------------------------------------------------------
__global__ __launch_bounds__(32) void s_accum_kernel(
    const float* __restrict__ u, const float* __restrict__ W,
    const float* __restrict__ c, float* __restrict__ s) {
  const int lane = threadIdx.x;
  const int m = lane & 15;
  const int hi = lane >> 4;
  const int b0 = blockIdx.x * 16;
  const int r0 = blockIdx.y * RC_SIZE;

  v8f acc[DIGITS] = {};
  const float* ub = u + (size_t)(b0 + m) * U_RSTRIDE + 2 * hi;

  for (int rr = 0; rr < RC_SIZE; ++rr) {
    const int r = r0 + rr;
    v2f a0 = *(const v2f*)(ub + (size_t)r * IN_CH);       // i = {2hi, 2hi+1}
    v2f a1 = *(const v2f*)(ub + (size_t)r * IN_CH + 4);   // i = {4+2hi, 5+2hi}
    const float* wr = W + (size_t)r * W_RSTRIDE + m * IN_CH + 2 * hi;
    const float* cr = c + r * DIGITS;
#pragma unroll
    for (int d = 0; d < DIGITS; ++d) {
      const float cv = cr[d];                             // scalar broadcast
      v2f wb0 = *(const v2f*)(wr + d * W_DSTRIDE);
      v2f wb1 = *(const v2f*)(wr + d * W_DSTRIDE + 4);
      wb0 *= cv;
      wb1 *= cv;
      acc[d] = __builtin_amdgcn_wmma_f32_16x16x4_f32(false, a0, false, wb0,
                                                     (short)0, acc[d], false, false);
      acc[d] = __builtin_amdgcn_wmma_f32_16x16x4_f32(false, a1, false, wb1,
                                                     (short)0, acc[d], false, false);
    }
  }
  // scatter partials: acc[d][j] -> s[b0 + j + 8*hi, d, m]
#pragma unroll
  for (int d = 0; d < DIGITS; ++d) {
    float* sp = s + (size_t)(b0 + 8 * hi) * S_BSTRIDE + d * OUT_CH + m;
#pragma unroll
    for (int j = 0; j < 8; ++j) atomicAdd(sp + (size_t)j * S_BSTRIDE, acc[d][j]);
  }
}

// ---------------------------------------------------------------------------
// sumsq = ||s||_F^2 (global norm of the whole (B,D,O) tensor, per reference)
// ---------------------------------------------------------------------------
__global__ void sumsq_kernel(const float* __restrict__ s,
                             float* __restrict__ sumsq) {
  int i = blockIdx.x * blockDim.x + threadIdx.x;
  float x = s[i];
  x *= x;
#pragma unroll
  for (int off = 16; off > 0; off >>= 1) x += __shfl_xor(x, off, 32);
  if ((threadIdx.x & 31) == 0) atomicAdd(sumsq, x);
}

// ---------------------------------------------------------------------------
// v = (sq/(1+sq)) * s / sqrt(sq)     (writes straight into d_out)
// ---------------------------------------------------------------------------
__global__ void squash_kernel(const float* __restrict__ s,
                              const float* __restrict__ sumsq,
                              float* __restrict__ vout) {
  int i = blockIdx.x * blockDim.x + threadIdx.x;
  float sq = *sumsq;
  float scale = (sq / (1.0f + sq)) * __frsqrt_rn(sq);
  vout[i] = s[i] * scale;
}

// ---------------------------------------------------------------------------
// b_logits[r,d] += sum_b sum_o u_hat[b,r,d,o] * v[b,d,o]
// One wave per (r, batch-quarter). W fragments preloaded once; u_hat tile is
// recomputed via WMMA, multiplied elementwise by the matching v fragment
// (identical C-layout), then fully reduced (VGPRs + wave32 butterfly).
// ---------------------------------------------------------------------------
__global__ __launch_bounds__(32) void agree_kernel(
    const float* __restrict__ u, const float* __restrict__ W,
    const float* __restrict__ v, float* __restrict__ blog) {
  const int lane = threadIdx.x;
  const int m = lane & 15;
  const int hi = lane >> 4;
  const int r = blockIdx.x;
  const int bt0 = blockIdx.y * BT_PER_BLOCK;

  v2f wb[DIGITS][2];
  const float* wr = W + (size_t)r * W_RSTRIDE + m * IN_CH + 2 * hi;
#pragma unroll
  for (int d = 0; d < DIGITS; ++d) {
    wb[d][0] = *(const v2f*)(wr + d * W_DSTRIDE);
    wb[d][1] = *(const v2f*)(wr + d * W_DSTRIDE + 4);
  }

  float bacc[DIGITS] = {};
  for (int bt = 0; bt < BT_PER_BLOCK; ++bt) {
    const int b0 = (bt0 + bt) * 16;
    const float* ub = u + (size_t)(b0 + m) * U_RSTRIDE + (size_t)r * IN_CH + 2 * hi;
    v2f a0 = *(const v2f*)(ub);
    v2f a1 = *(const v2f*)(ub + 4);
    const float* vb = v + (size_t)(b0 + 8 * hi) * S_BSTRIDE + m;
#pragma unroll
    for (int d = 0; d < DIGITS; ++d) {
      v8f frag = {};
      frag = __builtin_amdgcn_wmma_f32_16x16x4_f32(false, a0, false, wb[d][0],
                                                   (short)0, frag, false, false);
      frag = __builtin_amdgcn_wmma_f32_16x16x4_f32(false, a1, false, wb[d][1],
                                                   (short)0, frag, false, false);
      const float* vd = vb + d * OUT_CH;
      float p = 0.0f;
#pragma unroll
      for (int j = 0; j < 8; ++j) p += frag[j] * vd[(size_t)j * S_BSTRIDE];
      bacc[d] += p;
    }
  }
#pragma unroll
  for (int d = 0; d < DIGITS; ++d) {
    float x = bacc[d];
#pragma unroll
    for (int off = 16; off > 0; off >>= 1) x += __shfl_xor(x, off, 32);
    if (lane == 0) atomicAdd(blog + r * DIGITS + d, x);
  }
}

// ---------------------------------------------------------------------------
extern "C" void kernel_launch(void* const* d_in, const int* in_sizes, int n_in,
                              void* d_out, int out_size, void* d_ws, size_t ws_size,
                              hipStream_t stream) {
  const float* u = (const float*)d_in[0];   // (256, 1152, 8)
  const float* W = (const float*)d_in[1];   // (1, 1152, 10, 16, 8)
  float* vout = (float*)d_out;              // (256, 10, 16)

  float* c     = (float*)d_ws;                  // 11520
  float* blog  = c + R_NODES * DIGITS;          // 11520
  float* s     = blog + R_NODES * DIGITS;       // 40960
  float* sumsq = s + S_ELEMS;                   // 1

  hipMemsetAsync(blog, 0, R_NODES * DIGITS * sizeof(float), stream);

  for (int it = 0; it < NUM_ITER; ++it) {
    softmax_kernel<<<R_NODES / 128, 128, 0, stream>>>(blog, c);
    hipMemsetAsync(s, 0, S_ELEMS * sizeof(float), stream);
    hipMemsetAsync(sumsq, 0, sizeof(float), stream);
    s_accum_kernel<<<dim3(BATCH / 16, R_NODES / RC_SIZE), 32, 0, stream>>>(u, W, c, s);
    sumsq_kernel<<<S_ELEMS / 256, 256, 0, stream>>>(s, sumsq);
    squash_kernel<<<S_ELEMS / 256, 256, 0, stream>>>(s, sumsq, vout);
    if (it < NUM_ITER - 1) {
      agree_kernel<<<dim3(R_NODES, BATCH / (16 * BT_PER_BLOCK)), 32, 0, stream>>>(
          u, W, vout, blog);
    }
  }
}